// MyModel_87522843559533
// MI455X (gfx1250) — compile-verified
//
#include <hip/hip_runtime.h>

// mean(targets * logits) over N fp32 elements.
// Memory-bound: 2 x 128 MB streamed once -> ~11 us at 23.3 TB/s; FLOPs negligible.
// Pass 1: grid-stride, 4x-unrolled float4 NT loads (8 b128 loads in flight/wave),
//         4 independent FMA accumulators, wave reduction via
//         V_WMMA_F32_16X16X4_F32 with B = ones, block partial -> d_ws.
// Pass 2: single block reduces partials, scales by 1/N, writes d_out[0].

typedef __attribute__((ext_vector_type(2))) float v2f;
typedef __attribute__((ext_vector_type(4))) float v4f;
typedef __attribute__((ext_vector_type(8))) float v8f;

// Wave32 all-lane sum on the matrix pipe:
// A(16x4): lane L holds A[L%16][L<16?0:2] = x in VGPR0, VGPR1 = 0.
// B(4x16) = ones  ->  D[m][n] = sum_k A[m][k]  (every column identical).
// Per lane, sum of its 8 D VGPRs = sum over M=0..7 (lanes 0-15) or M=8..15
// (lanes 16-31). Lanes 0 and 16 hold the two half-sums; both spill to LDS.
__device__ __forceinline__ float block_reduce_wmma(float x, float* lds) {
    v2f a; a[0] = x;    a[1] = 0.0f;
    v2f b; b[0] = 1.0f; b[1] = 1.0f;
    v8f c = {};
    c = __builtin_amdgcn_wmma_f32_16x16x4_f32(
            /*neg_a=*/false, a, /*neg_b=*/false, b,
            /*c_mod=*/(short)0, c, /*reuse_a=*/false, /*reuse_b=*/false);
    float s = ((c[0] + c[1]) + (c[2] + c[3])) + ((c[4] + c[5]) + (c[6] + c[7]));

    const int lane = threadIdx.x & 31;
    const int wid  = threadIdx.x >> 5;
    if ((lane & 15) == 0)                 // lanes 0 and 16 of each wave
        lds[(wid << 1) | (lane >> 4)] = s;
    __syncthreads();
    float total = 0.0f;
    if (threadIdx.x == 0) {
        const int slots = (blockDim.x >> 5) << 1;   // 2 per wave
        for (int i = 0; i < slots; ++i) total += lds[i];
    }
    return total;   // valid in thread 0 only
}

__global__ void dot_partial_kernel(const float* __restrict__ t,
                                   const float* __restrict__ l,
                                   float* __restrict__ partial,
                                   long long n) {
    __shared__ float lds[16];

    const long long nvec   = n >> 2;                         // float4 count
    const long long stride = (long long)gridDim.x * blockDim.x;
    long long i = (long long)blockIdx.x * blockDim.x + threadIdx.x;

    const v4f* __restrict__ tv = (const v4f*)t;
    const v4f* __restrict__ lv = (const v4f*)l;

    v4f acc0 = {0.0f, 0.0f, 0.0f, 0.0f};
    v4f acc1 = {0.0f, 0.0f, 0.0f, 0.0f};
    v4f acc2 = {0.0f, 0.0f, 0.0f, 0.0f};
    v4f acc3 = {0.0f, 0.0f, 0.0f, 0.0f};

    // 4x unrolled: 8 independent global_load_b128 (TH=NT) in flight per pass
    for (; i + 3 * stride < nvec; i += 4 * stride) {
        v4f a0 = __builtin_nontemporal_load(tv + i);
        v4f a1 = __builtin_nontemporal_load(tv + i + stride);
        v4f a2 = __builtin_nontemporal_load(tv + i + 2 * stride);
        v4f a3 = __builtin_nontemporal_load(tv + i + 3 * stride);
        v4f b0 = __builtin_nontemporal_load(lv + i);
        v4f b1 = __builtin_nontemporal_load(lv + i + stride);
        v4f b2 = __builtin_nontemporal_load(lv + i + 2 * stride);
        v4f b3 = __builtin_nontemporal_load(lv + i + 3 * stride);
        acc0 += a0 * b0;
        acc1 += a1 * b1;
        acc2 += a2 * b2;
        acc3 += a3 * b3;
    }
    for (; i < nvec; i += stride) {
        v4f av = __builtin_nontemporal_load(tv + i);
        v4f bv = __builtin_nontemporal_load(lv + i);
        acc0 += av * bv;
    }

    acc0 += acc1;
    acc2 += acc3;
    acc0 += acc2;
    float x = (acc0[0] + acc0[1]) + (acc0[2] + acc0[3]);

    // scalar tail (n % 4), block 0 only; re-converges before the WMMA
    const long long tail0 = nvec << 2;
    if (blockIdx.x == 0) {
        for (long long j = tail0 + threadIdx.x; j < n; j += blockDim.x)
            x += t[j] * l[j];
    }

    float total = block_reduce_wmma(x, lds);
    if (threadIdx.x == 0) partial[blockIdx.x] = total;
}

__global__ void finalize_kernel(const float* __restrict__ partial, int P,
                                float* __restrict__ out, float invN) {
    __shared__ float lds[16];
    float x = 0.0f;
    for (int i = threadIdx.x; i < P; i += blockDim.x) x += partial[i];
    float total = block_reduce_wmma(x, lds);
    if (threadIdx.x == 0) out[0] = total * invN;
}

extern "C" void kernel_launch(void* const* d_in, const int* in_sizes, int n_in,
                              void* d_out, int out_size, void* d_ws, size_t ws_size,
                              hipStream_t stream) {
    const float* t   = (const float*)d_in[0];
    const float* l   = (const float*)d_in[1];
    float*       out = (float*)d_out;
    float*       ws  = (float*)d_ws;

    const long long n = (long long)in_sizes[0];

    const int threads = 256;                        // 8 wave32 per block
    int maxBlocks = 2048;                           // fills all WGPs many times over
    if (ws_size < (size_t)maxBlocks * sizeof(float))
        maxBlocks = (int)(ws_size / sizeof(float));
    if (maxBlocks < 1) maxBlocks = 1;

    const long long nvec = n >> 2;
    long long need = (nvec + threads - 1) / threads;
    if (need < 1) need = 1;
    const int blocks = (int)((need < (long long)maxBlocks) ? need : (long long)maxBlocks);

    dot_partial_kernel<<<blocks, threads, 0, stream>>>(t, l, ws, n);
    finalize_kernel<<<1, 256, 0, stream>>>(ws, blocks, out, 1.0f / (float)n);
}